// PathKeypointLoss_62199716380803
// MI455X (gfx1250) — compile-verified
//
#include <hip/hip_runtime.h>
#include <math.h>

typedef float v2f __attribute__((ext_vector_type(2)));
typedef float v8f __attribute__((ext_vector_type(8)));

#define B_SZ    32
#define N_PTS   4096
#define L_PATH  128
#define TILE_PTS 128

// ---------------- Kernel 1: per-batch sigma from pred min/max ----------------
__global__ void k1_sigma(const float* __restrict__ pred, float* __restrict__ sig)
{
    const int b = blockIdx.x;
    const int tid = threadIdx.x;
    float mnx = 3.4e38f, mxx = -3.4e38f, mny = 3.4e38f, mxy = -3.4e38f;
    for (int i = tid; i < N_PTS; i += 256) {
        const size_t gi = ((size_t)b * N_PTS + i) * 2;
        const float x = pred[gi], y = pred[gi + 1];
        mnx = fminf(mnx, x); mxx = fmaxf(mxx, x);
        mny = fminf(mny, y); mxy = fmaxf(mxy, y);
    }
    __shared__ float s0[256], s1[256], s2[256], s3[256];
    s0[tid] = mnx; s1[tid] = mxx; s2[tid] = mny; s3[tid] = mxy;
    __syncthreads();
    for (int s = 128; s > 0; s >>= 1) {
        if (tid < s) {
            s0[tid] = fminf(s0[tid], s0[tid + s]);
            s1[tid] = fmaxf(s1[tid], s1[tid + s]);
            s2[tid] = fminf(s2[tid], s2[tid + s]);
            s3[tid] = fmaxf(s3[tid], s3[tid + s]);
        }
        __syncthreads();
    }
    if (tid == 0) {
        const float rx = s1[0] - s0[0], ry = s3[0] - s2[0];
        sig[b * 4 + 0] = fmaxf(0.05f * rx, 1e-8f);  // sigma_line x
        sig[b * 4 + 1] = fmaxf(0.05f * ry, 1e-8f);  // sigma_line y
        sig[b * 4 + 2] = fmaxf(0.03f * rx, 1e-8f);  // sigma_point x
        sig[b * 4 + 3] = fmaxf(0.03f * ry, 1e-8f);  // sigma_point y
    }
}

// ------- Kernel 2: raw labels + cdist-min via V_WMMA_F32_16X16X4_F32 ---------
// grid = 32 batches * 32 tiles of 128 points; block = 256 = 8 waves * 16 pts.
__global__ void __launch_bounds__(256) k2_labels(
    const float* __restrict__ pred, const float* __restrict__ expert,
    const float* __restrict__ sig,
    float* __restrict__ pl_out, float* __restrict__ kl_out,
    float* __restrict__ dist_out)
{
    const int b    = blockIdx.x >> 5;
    const int tile = blockIdx.x & 31;
    const int tid  = threadIdx.x;
    const int lane = tid & 31;
    const int wave = tid >> 5;
    const int row  = lane & 15;
    const bool lo  = lane < 16;

    __shared__ float ex[128], ey[128];            // expert path
    __shared__ float ptx[128], pty[128];          // this tile's pred points
    __shared__ float segx[128], segy[128], il2[128], p0s[128], p0x[128], p0y[128];
    __shared__ float kxs[128], kys[128], ktm[128];
    __shared__ float etm[128];

    const float sLx = sig[b * 4 + 0], sLy = sig[b * 4 + 1];
    const float sPx = sig[b * 4 + 2], sPy = sig[b * 4 + 3];
    const float iLx = 1.f / sLx, iLy = 1.f / sLy;
    const float iPx2 = 1.f / (sPx * sPx), iPy2 = 1.f / (sPy * sPy);

    if (tid < 128) {
        ex[tid] = expert[((size_t)b * L_PATH + tid) * 2 + 0];
        ey[tid] = expert[((size_t)b * L_PATH + tid) * 2 + 1];
    } else {
        const int m = tid - 128;
        const size_t gi = ((size_t)b * N_PTS + tile * TILE_PTS + m) * 2;
        ptx[m] = pred[gi]; pty[m] = pred[gi + 1];
    }
    __syncthreads();
    if (tid < 128) {
        etm[tid] = ex[tid] * ex[tid] + ey[tid] * ey[tid];
        if (tid < 127) {
            const float sx = ex[tid + 1] - ex[tid], sy = ey[tid + 1] - ey[tid];
            const float l2 = fmaxf(sx * sx + sy * sy, 1e-12f);
            segx[tid] = sx; segy[tid] = sy; il2[tid] = 1.f / l2;
            p0s[tid] = ex[tid] * sx + ey[tid] * sy;
            p0x[tid] = ex[tid]; p0y[tid] = ey[tid];
        } else {  // pad segment 127: far away -> huge dist2, ignored by min
            segx[127] = 0.f; segy[127] = 0.f; il2[127] = 1.f; p0s[127] = 0.f;
            p0x[127] = 1e15f; p0y[127] = 1e15f;
        }
    } else {
        const int k = tid - 128;          // keypoint k = expert[k+1]
        if (k < 127) {
            const float kx = ex[k + 1], ky = ey[k + 1];
            kxs[k] = kx * iPx2; kys[k] = ky * iPy2;
            ktm[k] = kx * kx * iPx2 + ky * ky * iPy2;
        } else {  // pad keypoint: +1e30 constant term -> never the min
            kxs[127] = 0.f; kys[127] = 0.f; ktm[127] = 1e30f;
        }
    }
    __syncthreads();

    // A matrix (16 points x K=4; K=2,3 zero): VGPR0 = K0/K2, VGPR1 = K1/K3
    v2f A;
    A.x = lo ? ptx[wave * 16 + row] : 0.f;
    A.y = lo ? pty[wave * 16 + row] : 0.f;

    // Per-lane point registers: VGPR j / lane-half holds row M = j (+8 for hi)
    float px[8], py[8], pq[8], pt[8], mnL[8], mnK[8], mnC[8];
#pragma unroll
    for (int j = 0; j < 8; j++) {
        const int m = wave * 16 + j + (lo ? 0 : 8);
        px[j] = ptx[m]; py[j] = pty[m];
        pq[j] = px[j] * px[j] + py[j] * py[j];
        pt[j] = px[j] * px[j] * iPx2 + py[j] * py[j] * iPy2;
        mnL[j] = 3.4e38f; mnK[j] = 3.4e38f; mnC[j] = 3.4e38f;
    }

    const v8f zeroC = {0.f, 0.f, 0.f, 0.f, 0.f, 0.f, 0.f, 0.f};

    for (int t16 = 0; t16 < 8; ++t16) {
        const int n = t16 * 16 + row;   // this lane's column (segment/kp/path idx)
        v2f BL, BK, BC;                 // B: K rows striped across lane halves
        BL.x = lo ? segx[n] : 0.f;  BL.y = lo ? segy[n] : 0.f;
        BK.x = lo ? kxs[n]  : 0.f;  BK.y = lo ? kys[n]  : 0.f;
        BC.x = lo ? ex[n]   : 0.f;  BC.y = lo ? ey[n]   : 0.f;

        // D[m][n] = x_m * v_x[n] + y_m * v_y[n]
        v8f DL = __builtin_amdgcn_wmma_f32_16x16x4_f32(false, A, false, BL,
                                                       (short)0, zeroC, false, false);
        v8f DK = __builtin_amdgcn_wmma_f32_16x16x4_f32(false, A, false, BK,
                                                       (short)0, zeroC, false, false);
        v8f DC = __builtin_amdgcn_wmma_f32_16x16x4_f32(false, A, false, BC,
                                                       (short)0, zeroC, false, false);

        const float c_il2 = il2[n], c_p0s = p0s[n], c_sx = segx[n], c_sy = segy[n];
        const float c_px = p0x[n], c_py = p0y[n], c_ktm = ktm[n], c_etm = etm[n];
#pragma unroll
        for (int j = 0; j < 8; j++) {
            // line: t = clamp((pc.seg - p0.seg)/len2), proj, sigma-scaled dist2
            const float tp  = fminf(fmaxf((DL[j] - c_p0s) * c_il2, 0.f), 1.f);
            const float prx = c_px + tp * c_sx;
            const float pry = c_py + tp * c_sy;
            const float dx  = (px[j] - prx) * iLx;
            const float dy  = (py[j] - pry) * iLy;
            mnL[j] = fminf(mnL[j], dx * dx + dy * dy);
            // keypoint: |p|_s^2 + |k|_s^2 - 2 p.k_s
            mnK[j] = fminf(mnK[j], fmaxf(pt[j] + c_ktm - 2.f * DK[j], 0.f));
            // cdist: |p|^2 + |e|^2 - 2 p.e
            mnC[j] = fminf(mnC[j], fmaxf(pq[j] + c_etm - 2.f * DC[j], 0.f));
        }
    }

    // min over columns: reduce across the 16 lanes of each half (same M row)
#pragma unroll
    for (int j = 0; j < 8; j++) {
        for (int m = 8; m; m >>= 1) {
            mnL[j] = fminf(mnL[j], __shfl_xor(mnL[j], m, 16));
            mnK[j] = fminf(mnK[j], __shfl_xor(mnK[j], m, 16));
            mnC[j] = fminf(mnC[j], __shfl_xor(mnC[j], m, 16));
        }
    }
    if (row == 0) {
#pragma unroll
        for (int j = 0; j < 8; j++) {
            const int m = tile * TILE_PTS + wave * 16 + j + (lo ? 0 : 8);
            const size_t idx = (size_t)b * N_PTS + m;
            pl_out[idx]   = expf(-0.5f * mnL[j]);  // max exp == exp(-0.5*min d2)
            kl_out[idx]   = expf(-0.5f * mnK[j]);
            dist_out[idx] = sqrtf(mnC[j]);         // min sqrt == sqrt(min d2)
        }
    }
}

// ---------------- Kernel: per-batch sum (deterministic tree) -----------------
__global__ void k_batch_sum(const float* __restrict__ in, float* __restrict__ out)
{
    const int b = blockIdx.x, tid = threadIdx.x;
    float a = 0.f;
    for (int i = tid; i < N_PTS; i += 256) a += in[(size_t)b * N_PTS + i];
    __shared__ float sm[256];
    sm[tid] = a; __syncthreads();
    for (int s = 128; s > 0; s >>= 1) {
        if (tid < s) sm[tid] += sm[tid + s];
        __syncthreads();
    }
    if (tid == 0) out[b] = sm[0];
}

// -------- Kernel 4: sanitize (normalize + threshold) and combine -------------
__global__ void k4_combine(const float* __restrict__ pl, const float* __restrict__ kl,
                           const float* __restrict__ s_pl, const float* __restrict__ s_kl,
                           float* __restrict__ comb)
{
    const int b = blockIdx.x, tid = threadIdx.x;
    __shared__ float smx[2];
    if (tid == 0) {
        float mp = 0.f, mk = 0.f;
        for (int i = 0; i < B_SZ; i++) { mp = fmaxf(mp, s_pl[i]); mk = fmaxf(mk, s_kl[i]); }
        smx[0] = mp; smx[1] = mk;
    }
    __syncthreads();
    const bool np = smx[0] > 0.f, nk = smx[1] > 0.f;
    const float ip = 1.f / (s_pl[b] + 1e-8f), ik = 1.f / (s_kl[b] + 1e-8f);
    for (int i = tid; i < N_PTS; i += 256) {
        const size_t idx = (size_t)b * N_PTS + i;
        float p = pl[idx]; p = np ? p * ip : p; p = (p < 0.001f) ? 0.f : p;
        float q = kl[idx]; q = nk ? q * ik : q; q = (q < 0.001f) ? 0.f : q;
        comb[idx] = 1.0f * p + 1.5f * q;
    }
}

// ---- Kernel 6: per-batch align partial (with combined renorm) + smooth ------
__global__ void k6_align(const float* __restrict__ pred, const float* __restrict__ dist,
                         const float* __restrict__ comb, const float* __restrict__ c_sum,
                         float* __restrict__ a_part, float* __restrict__ s_part)
{
    const int b = blockIdx.x, tid = threadIdx.x;
    float aa = 0.f, ss = 0.f;
    for (int i = tid; i < N_PTS; i += 256) {
        const size_t idx = (size_t)b * N_PTS + i;
        aa += dist[idx] * comb[idx];
        if (i < N_PTS - 1) {
            const float dx = pred[(idx + 1) * 2 + 0] - pred[idx * 2 + 0];
            const float dy = pred[(idx + 1) * 2 + 1] - pred[idx * 2 + 1];
            ss += dx * dx + dy * dy;
        }
    }
    __shared__ float sma[256], sms[256];
    sma[tid] = aa; sms[tid] = ss; __syncthreads();
    for (int s = 128; s > 0; s >>= 1) {
        if (tid < s) { sma[tid] += sma[tid + s]; sms[tid] += sms[tid + s]; }
        __syncthreads();
    }
    if (tid == 0) {
        a_part[b] = sma[0] / (c_sum[b] + 1e-8f);
        s_part[b] = sms[0];
    }
}

// ---------------- Kernel 7: final scalars ------------------------------------
__global__ void k7_final(const float* __restrict__ a_part, const float* __restrict__ s_part,
                         float* __restrict__ out)
{
    if (threadIdx.x == 0 && blockIdx.x == 0) {
        float a = 0.f, s = 0.f;
        for (int i = 0; i < B_SZ; i++) { a += a_part[i]; s += s_part[i]; }
        const float align  = a / (float)(B_SZ * N_PTS);
        const float smooth = s / (float)(B_SZ * (N_PTS - 1) * 2);
        out[0] = align + 0.5f * smooth;
        out[1] = align;
        out[2] = smooth;
    }
}

extern "C" void kernel_launch(void* const* d_in, const int* in_sizes, int n_in,
                              void* d_out, int out_size, void* d_ws, size_t ws_size,
                              hipStream_t stream)
{
    const float* pred   = (const float*)d_in[0];  // (32,4096,2) f32
    const float* expert = (const float*)d_in[1];  // (32,128,2)  f32
    float* out = (float*)d_out;                   // 3 f32 scalars

    float* ws     = (float*)d_ws;
    float* sig    = ws;            // 32*4
    float* s_pl   = ws + 128;      // 32
    float* s_kl   = ws + 160;      // 32
    float* c_sum  = ws + 192;      // 32
    float* a_part = ws + 224;      // 32
    float* s_part = ws + 256;      // 32
    float* pl     = ws + 512;                   // 32*4096
    float* kl     = pl + B_SZ * N_PTS;          // 32*4096
    float* dist   = kl + B_SZ * N_PTS;          // 32*4096
    float* comb   = dist + B_SZ * N_PTS;        // 32*4096

    k1_sigma   <<<B_SZ, 256, 0, stream>>>(pred, sig);
    k2_labels  <<<B_SZ * 32, 256, 0, stream>>>(pred, expert, sig, pl, kl, dist);
    k_batch_sum<<<B_SZ, 256, 0, stream>>>(pl, s_pl);
    k_batch_sum<<<B_SZ, 256, 0, stream>>>(kl, s_kl);
    k4_combine <<<B_SZ, 256, 0, stream>>>(pl, kl, s_pl, s_kl, comb);
    k_batch_sum<<<B_SZ, 256, 0, stream>>>(comb, c_sum);
    k6_align   <<<B_SZ, 256, 0, stream>>>(pred, dist, comb, c_sum, a_part, s_part);
    k7_final   <<<1, 32, 0, stream>>>(a_part, s_part, out);
}